// UIModel_50646254355232
// MI455X (gfx1250) — compile-verified
//
#include <hip/hip_runtime.h>
#include <cstdint>

#define NUM_USERS 100000
#define NUM_ITEMS 50000
#define NN (NUM_USERS + NUM_ITEMS)
#define EMB 64
#define NUM_LAYERS 3
#define TILE 512

// ---------------- CDNA5 async global->LDS helpers ----------------
// GLOBAL_LOAD_ASYNC_TO_LDS_B32/B128: per-lane copy from global memory directly
// into LDS, tracked by ASYNCcnt (no VGPR round trip).
__device__ __forceinline__ void async_g2lds_b32(void* lds_dst, const void* gsrc) {
  uint32_t loff = (uint32_t)(uintptr_t)lds_dst;   // low 32 bits of generic shared addr == LDS byte offset
  uint64_t ga   = (uint64_t)(uintptr_t)gsrc;
  asm volatile("global_load_async_to_lds_b32 %0, %1, off"
               :: "v"(loff), "v"(ga) : "memory");
}
__device__ __forceinline__ void async_g2lds_b128(void* lds_dst, const void* gsrc) {
  uint32_t loff = (uint32_t)(uintptr_t)lds_dst;
  uint64_t ga   = (uint64_t)(uintptr_t)gsrc;
  asm volatile("global_load_async_to_lds_b128 %0, %1, off"
               :: "v"(loff), "v"(ga) : "memory");
}
__device__ __forceinline__ void wait_asynccnt0() {
#if __has_builtin(__builtin_amdgcn_s_wait_asynccnt)
  __builtin_amdgcn_s_wait_asynccnt(0);
#else
  asm volatile("s_wait_asynccnt 0x0" ::: "memory");
#endif
}

// Hardware FP32 atomic add (global_atomic_add_f32, non-returning at L2),
// avoiding any CAS-loop legalization of plain atomicAdd(float*).
__device__ __forceinline__ void atomic_add_f32(float* p, float v) {
  unsafeAtomicAdd(p, v);
}

// ---------------- hot kernel first (so the disasm snippet shows it) ----------
// One block = one tile of TILE undirected edges. Edge metadata (u, i, w) is
// streamed into LDS via async global->LDS DMA (b128 body + b32 tail), then
// each wave32 handles one edge per iteration: half-wave 0 does u->i, half-wave
// 1 does i->u; 16 lanes x float4 = one 256B embedding row per half-wave, so a
// single global_load_b128 per wave covers both gather rows of the edge.
__global__ void k_propagate(const int* __restrict__ eu, const int* __restrict__ ei,
                            const float* __restrict__ wgt,
                            const float* __restrict__ x, float* __restrict__ y, int E) {
  __shared__ alignas(16) int   s_u[TILE];
  __shared__ alignas(16) int   s_i[TILE];
  __shared__ alignas(16) float s_w[TILE];

  const int base = blockIdx.x * TILE;
  int count = E - base;
  if (count <= 0) return;
  if (count > TILE) count = TILE;

  const int c4 = count & ~3;
  for (int t = (threadIdx.x << 2); t < c4; t += (blockDim.x << 2)) {
    async_g2lds_b128(&s_u[t], &eu[base + t]);
    async_g2lds_b128(&s_i[t], &ei[base + t]);
    async_g2lds_b128(&s_w[t], &wgt[base + t]);
  }
  for (int t = c4 + threadIdx.x; t < count; t += blockDim.x) {
    async_g2lds_b32(&s_u[t], &eu[base + t]);
    async_g2lds_b32(&s_i[t], &ei[base + t]);
    async_g2lds_b32(&s_w[t], &wgt[base + t]);
  }
  wait_asynccnt0();      // my wave's async DMAs have landed in LDS
  __syncthreads();       // everyone's have

  const int lane = threadIdx.x & 31;
  const int half = lane >> 4;          // 0: u -> i, 1: i -> u
  const int sub  = lane & 15;          // float4 slot within the 64-float row
  const int wv   = threadIdx.x >> 5;
  const int nw   = blockDim.x >> 5;

  for (int t = wv; t < count; t += nw) {
    const int   u = s_u[t];
    const int   v = NUM_USERS + s_i[t];
    const float w = s_w[t];
    const int srow = half ? v : u;     // all offsets fit in 32 bits (x/y = 38.4MB)
    const int drow = half ? u : v;

    float4 a = ((const float4*)x)[srow * (EMB / 4) + sub];
    float* d = y + drow * EMB + 4 * sub;
    atomic_add_f32(d + 0, w * a.x);
    atomic_add_f32(d + 1, w * a.y);
    atomic_add_f32(d + 2, w * a.z);
    atomic_add_f32(d + 3, w * a.w);
  }
}

// ---------------- remaining kernels ----------------
__global__ void k_zero(float* __restrict__ p, long n) {          // small scalar zero
  long i = (long)blockIdx.x * blockDim.x + threadIdx.x;
  long stride = (long)gridDim.x * blockDim.x;
  for (; i < n; i += stride) p[i] = 0.0f;
}

__global__ void k_zero4(float4* __restrict__ p, long n4) {       // bulk b128 zero
  long i = (long)blockIdx.x * blockDim.x + threadIdx.x;
  if (i >= n4) return;
  p[i] = make_float4(0.0f, 0.0f, 0.0f, 0.0f);
}

__global__ void k_degree(const int* __restrict__ eu, const int* __restrict__ ei,
                         int* __restrict__ deg, int E) {
  int e = blockIdx.x * blockDim.x + threadIdx.x;
  if (e >= E) return;
  atomicAdd(&deg[eu[e]], 1);
  atomicAdd(&deg[NUM_USERS + ei[e]], 1);
}

__global__ void k_invsqrt(float* __restrict__ dsq, int n) {
  int i = blockIdx.x * blockDim.x + threadIdx.x;
  if (i >= n) return;
  int d = ((const int*)dsq)[i];
  dsq[i] = (d > 0) ? rsqrtf((float)d) : 0.0f;
}

__global__ void k_weights(const int* __restrict__ eu, const int* __restrict__ ei,
                          const float* __restrict__ dsq, float* __restrict__ wgt, int E) {
  int e = blockIdx.x * blockDim.x + threadIdx.x;
  if (e >= E) return;
  wgt[e] = dsq[eu[e]] * dsq[NUM_USERS + ei[e]];
}

__global__ void k_init(const float4* __restrict__ ue, const float4* __restrict__ ie,
                       float4* __restrict__ x, float4* __restrict__ out) {
  long i = (long)blockIdx.x * blockDim.x + threadIdx.x;
  const long uN4 = (long)NUM_USERS * EMB / 4;
  const long tN4 = (long)NN * EMB / 4;
  if (i >= tN4) return;
  if (i < uN4) {
    float4 v = ue[i];
    x[i] = v;
    out[i] = make_float4(0.25f * v.x, 0.25f * v.y, 0.25f * v.z, 0.25f * v.w);
  } else {
    x[i] = ie[i - uN4];
  }
}

__global__ void k_accum(const float4* __restrict__ y, float4* __restrict__ out) {
  long i = (long)blockIdx.x * blockDim.x + threadIdx.x;
  const long uN4 = (long)NUM_USERS * EMB / 4;
  if (i >= uN4) return;
  float4 a = y[i];
  float4 o = out[i];
  o.x += 0.25f * a.x; o.y += 0.25f * a.y;
  o.z += 0.25f * a.z; o.w += 0.25f * a.w;
  out[i] = o;
}

// ---------------- launch ----------------
extern "C" void kernel_launch(void* const* d_in, const int* in_sizes, int n_in,
                              void* d_out, int out_size, void* d_ws, size_t ws_size,
                              hipStream_t stream) {
  const int*   eu = (const int*)d_in[0];
  const int*   ei = (const int*)d_in[1];
  const float* ue = (const float*)d_in[2];
  const float* ie = (const float*)d_in[3];
  float*       out = (float*)d_out;
  const int    E  = in_sizes[0];

  // workspace layout (floats): x[N*64] | y[N*64] | w[E] | deg/invsqrt[N]
  float* x   = (float*)d_ws;
  float* y   = x + (size_t)NN * EMB;
  float* wgt = y + (size_t)NN * EMB;
  float* dsq = wgt + (size_t)E;

  const long tot  = (long)NN * EMB;          // 9.6M floats
  const long tot4 = tot / 4;
  const long uN4  = (long)NUM_USERS * EMB / 4;
  const int  gE   = (E + 255) / 256;
  const int  gT   = (E + TILE - 1) / TILE;

  // degrees (integer atomics into dsq's storage; bit pattern 0 == 0.0f)
  k_zero<<<256, 256, 0, stream>>>(dsq, NN);
  k_degree<<<gE, 256, 0, stream>>>(eu, ei, (int*)dsq, E);
  k_invsqrt<<<(NN + 255) / 256, 256, 0, stream>>>(dsq, NN);
  k_weights<<<gE, 256, 0, stream>>>(eu, ei, dsq, wgt, E);

  k_init<<<(int)((tot4 + 255) / 256), 256, 0, stream>>>(
      (const float4*)ue, (const float4*)ie, (float4*)x, (float4*)out);

  float* cx = x;
  float* cy = y;
  for (int l = 0; l < NUM_LAYERS; ++l) {
    k_zero4<<<(int)((tot4 + 255) / 256), 256, 0, stream>>>((float4*)cy, tot4);
    k_propagate<<<gT, 256, 0, stream>>>(eu, ei, wgt, cx, cy, E);
    k_accum<<<(int)((uN4 + 255) / 256), 256, 0, stream>>>((const float4*)cy, (float4*)out);
    float* t = cx; cx = cy; cy = t;
  }
}